// STL_70059506532519
// MI455X (gfx1250) — compile-verified
//
#include <hip/hip_runtime.h>

typedef float v2f __attribute__((ext_vector_type(2)));
typedef float v4f __attribute__((ext_vector_type(4)));
typedef float v8f __attribute__((ext_vector_type(8)));

#define NUM_HEADS 8
#define HEAD_DIM 64
#define TOKEN_NUM 10
#define EMBED_DIMS 512
#define QUERY_DIM 256
#define BATCH 131072

// ws float layout:
//   [0, 20480)        A_swz : stage-1 B operands. [ks(64)][nt(5)] -> 64 floats,
//                     entry = Ahat[4*ks + v + (lane>=16?2:0)][16*nt + (lane&15)] at word 2*lane+v
//   [20480, 21248)    K_swz : stage-2 B operands. [ks2(3)][dt(4)] -> 64 floats,
//                     entry = keys_raw[4*ks2 + v + (lane>=16?2:0)][16*dt + (lane&15)] (0 if t>=10)
#define A_SWZ_FLOATS 20480
#define K_SWZ_FLOATS 768

__global__ __launch_bounds__(256) void prep_kernel(const float* __restrict__ embed,
                                                   const float* __restrict__ Wq,
                                                   const float* __restrict__ Wk,
                                                   float* __restrict__ ws) {
    __shared__ float skr[TOKEN_NUM * HEAD_DIM];    // keys_raw = tanh(embed), 640 f
    __shared__ float skk[TOKEN_NUM * EMBED_DIMS];  // k = keys_raw @ Wk.T,   5120 f
    const int tid = threadIdx.x;

    for (int i = tid; i < TOKEN_NUM * HEAD_DIM; i += 256) skr[i] = tanhf(embed[i]);
    __syncthreads();

    // kk[t][e] = sum_d skr[t][d] * Wk[e][d]
    for (int e = tid; e < EMBED_DIMS; e += 256) {
        float s[TOKEN_NUM];
        #pragma unroll
        for (int t = 0; t < TOKEN_NUM; ++t) s[t] = 0.f;
        for (int d = 0; d < HEAD_DIM; ++d) {
            float w = Wk[e * HEAD_DIM + d];
            #pragma unroll
            for (int t = 0; t < TOKEN_NUM; ++t) s[t] += w * skr[t * HEAD_DIM + d];
        }
        for (int t = 0; t < TOKEN_NUM; ++t) skk[t * EMBED_DIMS + e] = s[t];
    }
    __syncthreads();

    // Ahat[c][h*10+t] = (1/8) * sum_d Wq[h*64+d][c] * kk[t][h*64+d], scattered into
    // WMMA-B swizzled layout. Thread tid owns k-row c = tid (exactly 256 threads).
    {
        const int c = tid;
        const int r = c & 3, ks = c >> 2;
        const int v = r & 1, hiadd = (r >> 1) << 4;  // +16 lanes when c%4 >= 2
        for (int h = 0; h < NUM_HEADS; ++h) {
            float s[TOKEN_NUM];
            #pragma unroll
            for (int t = 0; t < TOKEN_NUM; ++t) s[t] = 0.f;
            for (int d = 0; d < HEAD_DIM; ++d) {
                float w = Wq[(h * HEAD_DIM + d) * QUERY_DIM + c];
                #pragma unroll
                for (int t = 0; t < TOKEN_NUM; ++t)
                    s[t] += w * skk[t * EMBED_DIMS + h * HEAD_DIM + d];
            }
            for (int t = 0; t < TOKEN_NUM; ++t) {
                int j = h * TOKEN_NUM + t;
                int nt = j >> 4;
                int l = (j & 15) + hiadd;
                ws[(ks * 5 + nt) * 64 + 2 * l + v] = 0.125f * s[t];
            }
        }
    }

    // K_swz: stage-2 B operands (t padded to 12 with zeros)
    for (int i = tid; i < K_SWZ_FLOATS; i += 256) {
        int v = i & 1, l = (i >> 1) & 31, tile = i >> 6;
        int ks2 = tile >> 2, dt = tile & 3;
        int t = 4 * ks2 + v + ((l >= 16) ? 2 : 0);
        int d = 16 * dt + (l & 15);
        ws[A_SWZ_FLOATS + i] = (t < TOKEN_NUM) ? skr[t * HEAD_DIM + d] : 0.f;
    }
}

__global__ __launch_bounds__(256) void attn_kernel(const float* __restrict__ X,
                                                   const float* __restrict__ ws,
                                                   float* __restrict__ style,
                                                   float* __restrict__ scores) {
    __shared__ float sA[A_SWZ_FLOATS];  // 80 KB: stage-1 B operands
    __shared__ float sS[8][1600];       // per-wave 16 rows x 100 words (head-padded scores)

    const int tid = threadIdx.x;
    const int wave = tid >> 5;
    const int lane = tid & 31;
    const int hi16 = lane >> 4;   // 0: lanes 0-15, 1: lanes 16-31
    const int lm   = lane & 15;
    const int vrowoff = hi16 << 3;

    // Stage Ahat into LDS (float4-coalesced)
    {
        const v4f* src = (const v4f*)ws;
        v4f* dst = (v4f*)sA;
        for (int i = tid; i < A_SWZ_FLOATS / 4; i += 256) dst[i] = src[i];
    }
    // Stage-2 key B-operands held in registers for the whole kernel (24 VGPRs)
    v2f kb[3][4];
    #pragma unroll
    for (int ks2 = 0; ks2 < 3; ++ks2)
        #pragma unroll
        for (int dt = 0; dt < 4; ++dt)
            kb[ks2][dt] = *(const v2f*)(ws + A_SWZ_FLOATS + (ks2 * 4 + dt) * 64 + 2 * lane);

    // Zero the t=10,11 pad slots of this wave's score buffer (wave-private region)
    float* sSw = &sS[wave][0];
    #pragma unroll
    for (int i = 0; i < 8; ++i) {
        int s = lane + 32 * i;          // 0..255
        int r = s >> 4, p = s & 15;     // row, head*2 + slot
        sSw[r * 100 + (p >> 1) * 12 + 10 + (p & 1)] = 0.f;
    }
    __syncthreads();

    const int rowbase = blockIdx.x * 256 + wave * 32;

    // ---------- Stage 1: S(32x80) = X(32x256) @ Ahat(256x80), f32 WMMA ----------
    v8f acc[2][5];
    #pragma unroll
    for (int m = 0; m < 2; ++m)
        #pragma unroll
        for (int nt = 0; nt < 5; ++nt)
            #pragma unroll
            for (int e = 0; e < 8; ++e) acc[m][nt][e] = 0.f;

    // A-operand address: lane lm holds row (rowbase+16m+lm); lanes>=16 read cols k+2,k+3
    const float* x0 = X + (size_t)(rowbase + lm) * QUERY_DIM + 2 * hi16;
    const float* x1 = x0 + 16 * QUERY_DIM;

    #pragma unroll 4
    for (int ks = 0; ks < 64; ++ks) {
        v2f a0 = *(const v2f*)(x0 + 4 * ks);
        v2f a1 = *(const v2f*)(x1 + 4 * ks);
        #pragma unroll
        for (int nt = 0; nt < 5; ++nt) {
            v2f b = *(const v2f*)(sA + (ks * 5 + nt) * 64 + 2 * lane);
            acc[0][nt] = __builtin_amdgcn_wmma_f32_16x16x4_f32(
                false, a0, false, b, (short)0, acc[0][nt], false, false);
            acc[1][nt] = __builtin_amdgcn_wmma_f32_16x16x4_f32(
                false, a1, false, b, (short)0, acc[1][nt], false, false);
        }
    }

    // ---------- Epilogue per m-tile: emit scores, then Stage 2 via WMMA ----------
    #pragma unroll
    for (int m = 0; m < 2; ++m) {
        // Scatter S tile: to global (h-major layout) and to LDS (head-padded, stride 100)
        #pragma unroll
        for (int nt = 0; nt < 5; ++nt) {
            int c = nt * 16 + lm;       // 0..79, always valid (80 = 5*16)
            int h = c / 10;
            int t = c - 10 * h;
            #pragma unroll
            for (int v = 0; v < 8; ++v) {
                float val = acc[m][nt][v];
                int r = v + vrowoff;
                sSw[r * 100 + h * 12 + t] = val;
                int grow = rowbase + 16 * m + r;
                scores[h * (BATCH * TOKEN_NUM) + grow * TOKEN_NUM + t] = val;
            }
        }

        // Stage 2: out[16x64] per head = S_h(16x12) @ keys(12x64), t padded to 12
        #pragma unroll
        for (int h = 0; h < NUM_HEADS; ++h) {
            v8f o[4];
            #pragma unroll
            for (int dt = 0; dt < 4; ++dt)
                #pragma unroll
                for (int e = 0; e < 8; ++e) o[dt][e] = 0.f;
            #pragma unroll
            for (int ks2 = 0; ks2 < 3; ++ks2) {
                // A-operand: lane lm = row, lanes>=16 read t+2,t+3 (bank-conflict-free, stride 100)
                v2f a = *(const v2f*)(sSw + lm * 100 + h * 12 + 4 * ks2 + 2 * hi16);
                #pragma unroll
                for (int dt = 0; dt < 4; ++dt)
                    o[dt] = __builtin_amdgcn_wmma_f32_16x16x4_f32(
                        false, a, false, kb[ks2][dt], (short)0, o[dt], false, false);
            }
            #pragma unroll
            for (int dt = 0; dt < 4; ++dt)
                #pragma unroll
                for (int v = 0; v < 8; ++v) {
                    int r = v + vrowoff;
                    int grow = rowbase + 16 * m + r;
                    style[(size_t)grow * EMBED_DIMS + h * HEAD_DIM + dt * 16 + lm] = o[dt][v];
                }
        }
    }
}

extern "C" void kernel_launch(void* const* d_in, const int* in_sizes, int n_in,
                              void* d_out, int out_size, void* d_ws, size_t ws_size,
                              hipStream_t stream) {
    (void)in_sizes; (void)n_in; (void)out_size; (void)ws_size;
    const float* inputs = (const float*)d_in[0];
    const float* embed  = (const float*)d_in[1];
    const float* Wq     = (const float*)d_in[2];
    const float* Wk     = (const float*)d_in[3];
    float* out    = (float*)d_out;
    float* style  = out;                                  // (131072, 512)
    float* scores = out + (size_t)BATCH * EMBED_DIMS;     // (8, 131072, 1, 10)
    float* ws     = (float*)d_ws;

    prep_kernel<<<1, 256, 0, stream>>>(embed, Wq, Wk, ws);
    attn_kernel<<<BATCH / 256, 256, 0, stream>>>(inputs, ws, style, scores);
}